// LagrangianDynamics_88742614270197
// MI455X (gfx1250) — compile-verified
//
#include <hip/hip_runtime.h>

#define NL 6
#define BLOCK 256
#define TILE (BLOCK * NL)   // 1536 doubles per array per block

// ---- CDNA5 async global->LDS path (ASYNCcnt-tracked) -----------------------
#if defined(__HIP_DEVICE_COMPILE__) && __has_builtin(__builtin_amdgcn_global_load_async_to_lds_b128)
#define USE_ASYNC_LDS 1
#else
#define USE_ASYNC_LDS 0
#endif

typedef int async_v4i __attribute__((vector_size(4 * sizeof(int))));

static __device__ __forceinline__ void async_copy_b128(const double* gsrc, double* ldst) {
#if USE_ASYNC_LDS
  __builtin_amdgcn_global_load_async_to_lds_b128(
      (__attribute__((address_space(1))) async_v4i*)(gsrc),
      (__attribute__((address_space(3))) async_v4i*)(ldst),
      /*offset=*/0, /*cpol=*/0);
#else
  ldst[0] = gsrc[0];
  ldst[1] = gsrc[1];
#endif
}

static __device__ __forceinline__ void wait_async_lds() {
#if USE_ASYNC_LDS
#if __has_builtin(__builtin_amdgcn_s_wait_asynccnt)
  __builtin_amdgcn_s_wait_asynccnt(0);
#else
  asm volatile("s_wait_asynccnt 0" ::: "memory");
#endif
#endif
}

// ---- closed-form mass matrix: M[i][j] = (XiXj + YiYj) * W[max(i,j)] + diag(I22)
static __device__ __forceinline__ void mass_mat(
    const double s[NL], const double c[NL],
    const double comx[NL], const double W[NL], const double I22[NL],
    double M[NL][NL])
{
  const double LINKL[NL] = {0.5, 0.4, 0.3, 0.25, 0.2, 0.15};
  double SA = 0.0, SS = 0.0, CA = 0.0, CS = 0.0;
  double X[NL], Y[NL];
#pragma unroll
  for (int l = 0; l < NL; ++l) {
    SA += LINKL[l] * s[l];  SS += s[l];
    CA += LINKL[l] * c[l];  CS += c[l];
    X[l] = SA + comx[l] * SS;   // sign of Jx squares away
    Y[l] = CA + comx[l] * CS;
  }
#pragma unroll
  for (int i = 0; i < NL; ++i) {
#pragma unroll
    for (int j = i; j < NL; ++j) {
      double v = (X[i] * X[j] + Y[i] * Y[j]) * W[j];  // j == max(i,j)
      if (i == j) v += I22[i];
      M[i][j] = v;
      M[j][i] = v;
    }
  }
}

__global__ __launch_bounds__(BLOCK)
void lag_dyn_kernel(const double* __restrict__ q,
                    const double* __restrict__ qd,
                    const double* __restrict__ tau,
                    const double* __restrict__ link_masses,
                    const double* __restrict__ com_positions,
                    const double* __restrict__ inertia_params,
                    const double* __restrict__ fric,
                    const double* __restrict__ damp,
                    double* __restrict__ out,
                    int B)
{
  __shared__ double sm[3 * TILE];   // 36 KB of the 320 KB WGP pool

  const int tid = threadIdx.x;
  const int b0  = blockIdx.x * BLOCK;
  const int lim = B * NL;

  // ---- stage q/qd/tau tiles into LDS (coalesced b128 lanes, async engine) --
  {
    const double* bases[3] = { q, qd, tau };
#pragma unroll
    for (int a = 0; a < 3; ++a) {
      const double* g = bases[a] + (size_t)b0 * NL;
      double*       l = &sm[a * TILE];
#pragma unroll
      for (int ck = 0; ck < 3; ++ck) {
        int e = ck * (BLOCK * 2) + tid * 2;      // 2 doubles / lane / op
        if (b0 * NL + e + 1 < lim) async_copy_b128(g + e, l + e);
      }
    }
  }
  wait_async_lds();
  __syncthreads();

  const int b = b0 + tid;
  if (b >= B) return;

  double qv[NL], qdv[NL], tv[NL];
#pragma unroll
  for (int i = 0; i < NL; ++i) {
    qv[i]  = sm[            tid * NL + i];
    qdv[i] = sm[    TILE + tid * NL + i];
    tv[i]  = sm[2 * TILE + tid * NL + i];
  }

  // ---- uniform per-link parameters (scalarized by compiler) ----------------
  double comx[NL], I22[NL], W[NL], fc[NL], dc[NL];
#pragma unroll
  for (int i = 0; i < NL; ++i) {
    comx[i] = com_positions[i * 3];        // com[:,0]
    I22[i]  = inertia_params[i * 6 + 5];   // inertia[:,5]
    fc[i]   = fric[i];
    dc[i]   = damp[i];
  }
  {
    double acc = 0.0;
#pragma unroll
    for (int i = NL - 1; i >= 0; --i) { acc += link_masses[i]; W[i] = acc; }
  }

  // ---- base angles: the ONLY transcendentals for all 7 mass evaluations ---
  double s[NL], c[NL];
  {
    double ang = 0.0;
#pragma unroll
    for (int i = 0; i < NL; ++i) { ang += qv[i]; sincos(ang, &s[i], &c[i]); }
  }

  double M0[NL][NL];
  mass_mat(s, c, comx, W, I22, M0);

  // ---- Coriolis via finite differences, exact angle-addition for +EPS -----
  const double EPSV = 1e-6;
  const double se = 9.99999999999999983e-07;  // rounds to the double 1e-6, ~sin(1e-6)
  const double ce = 0.9999999999995;          // cos(1e-6), correctly rounded
  const double half_over_eps = 0.5 / EPSV;

  double Cm[NL][NL];
#pragma unroll
  for (int i = 0; i < NL; ++i)
#pragma unroll
    for (int j = 0; j < NL; ++j) Cm[i][j] = 0.0;

#pragma unroll
  for (int k = 0; k < NL; ++k) {
    double sp[NL], cp[NL];
#pragma unroll
    for (int i = 0; i < NL; ++i) {
      if (i >= k) { sp[i] = s[i] * ce + c[i] * se; cp[i] = c[i] * ce - s[i] * se; }
      else        { sp[i] = s[i];                  cp[i] = c[i]; }
    }
    double Mk[NL][NL];
    mass_mat(sp, cp, comx, W, I22, Mk);
    const double coef = half_over_eps * qdv[k];
#pragma unroll
    for (int i = 0; i < NL; ++i)
#pragma unroll
      for (int j = i; j < NL; ++j) {
        double d = coef * (Mk[i][j] - M0[i][j]);
        Cm[i][j] += d;
        if (i != j) Cm[j][i] += d;
      }
  }

  // ---- rhs = tau - C*qd - friction ----------------------------------------
  double rhs[NL];
#pragma unroll
  for (int i = 0; i < NL; ++i) {
    double cq = 0.0;
#pragma unroll
    for (int j = 0; j < NL; ++j) cq += Cm[i][j] * qdv[j];
    double fr = fc[i] * tanh(qdv[i] * 10.0) + dc[i] * qdv[i];
    rhs[i] = tv[i] - cq - fr;
  }

  // ---- Cholesky solve (M0 SPD), fully register-resident --------------------
  double L[NL][NL], invd[NL];
#pragma unroll
  for (int i = 0; i < NL; ++i) {
#pragma unroll
    for (int j = 0; j <= i; ++j) {
      double sum = M0[i][j];
#pragma unroll
      for (int p = 0; p < j; ++p) sum -= L[i][p] * L[j][p];
      if (i == j) { double dsq = sqrt(sum); L[i][i] = dsq; invd[i] = 1.0 / dsq; }
      else        { L[i][j] = sum * invd[j]; }
    }
  }
  double yv[NL];
#pragma unroll
  for (int i = 0; i < NL; ++i) {
    double t = rhs[i];
#pragma unroll
    for (int p = 0; p < i; ++p) t -= L[i][p] * yv[p];
    yv[i] = t * invd[i];
  }
  double xv[NL];
#pragma unroll
  for (int i = NL - 1; i >= 0; --i) {
    double t = yv[i];
#pragma unroll
    for (int p = i + 1; p < NL; ++p) t -= L[p][i] * xv[p];
    xv[i] = t * invd[i];
  }

#pragma unroll
  for (int i = 0; i < NL; ++i) out[(size_t)b * NL + i] = xv[i];
}

extern "C" void kernel_launch(void* const* d_in, const int* in_sizes, int n_in,
                              void* d_out, int out_size, void* d_ws, size_t ws_size,
                              hipStream_t stream) {
  const double* q   = (const double*)d_in[0];
  const double* qd  = (const double*)d_in[1];
  const double* tau = (const double*)d_in[2];
  const double* lm  = (const double*)d_in[3];
  const double* com = (const double*)d_in[4];
  const double* ine = (const double*)d_in[5];
  const double* fr  = (const double*)d_in[6];
  const double* da  = (const double*)d_in[7];
  double* out = (double*)d_out;

  const int B = in_sizes[0] / NL;
  dim3 grid((B + BLOCK - 1) / BLOCK), block(BLOCK);
  lag_dyn_kernel<<<grid, block, 0, stream>>>(q, qd, tau, lm, com, ine, fr, da, out, B);
}